// InceptionBlock_41343355191554
// MI455X (gfx1250) — compile-verified
//
#include <hip/hip_runtime.h>
#include <hip/hip_bf16.h>

typedef __attribute__((ext_vector_type(2))) float v2f;
typedef __attribute__((ext_vector_type(8))) float v8f;

#define N_NODES 100000
#define D 64
#define N_EDGES 1250000

// ---------------------------------------------------------------------------
// GEMM: Out[M x 64] = X[M x 64] * W[64 x 64] (+ bias) using V_WMMA_F32_16X16X4_F32
// One wave per 16x16 output tile; K=64 as 16 chained K=4 WMMA ops (f32 exact).
// ---------------------------------------------------------------------------
__global__ void gemm_wmma_f32(const float* __restrict__ X,
                              const float* __restrict__ W,
                              const float* __restrict__ bias,
                              float* __restrict__ Out, int M) {
  const int wave = threadIdx.x >> 5;
  const int lane = threadIdx.x & 31;
  const int tile = blockIdx.x * (blockDim.x >> 5) + wave;
  const int mt = tile >> 2;       // 4 N-tiles (64/16)
  const int nt = tile & 3;
  if (mt * 16 >= M) return;
  const int m0 = mt * 16;
  const int n0 = nt * 16;
  const int lm = lane & 15;        // M (for A) / N (for B,C,D) within tile
  const int lk = (lane >> 4) * 2;  // K sub-offset: lanes 0-15 -> K+{0,1}, 16-31 -> K+{2,3}

  const float* xrow = X + (size_t)(m0 + lm) * D;
  v8f c = {};
#pragma unroll
  for (int k0 = 0; k0 < D; k0 += 4) {
    v2f a, b;
    const float2 av = *(const float2*)(xrow + k0 + lk);   // 8B-aligned (k0+lk even)
    a.x = av.x;
    a.y = av.y;
    b.x = W[(size_t)(k0 + lk) * D + n0 + lm];
    b.y = W[(size_t)(k0 + lk + 1) * D + n0 + lm];
    c = __builtin_amdgcn_wmma_f32_16x16x4_f32(false, a, false, b, (short)0, c,
                                              false, false);
  }

  const float bcol = bias ? bias[n0 + lm] : 0.0f;
  const int rbase = m0 + (lane >> 4) * 8;   // C/D: VGPR r holds M = r + 8*(lane>=16)
#pragma unroll
  for (int r = 0; r < 8; ++r) {
    Out[(size_t)(rbase + r) * D + n0 + lm] = c[r] + bcol;
  }
}

// ---------------------------------------------------------------------------
// Degree / norm helpers
// ---------------------------------------------------------------------------
__global__ void fill_f32(float* __restrict__ p, float v, int n) {
  int i = blockIdx.x * blockDim.x + threadIdx.x;
  if (i < n) p[i] = v;
}

__global__ void degree_kernel(const int* __restrict__ dst, float* __restrict__ deg,
                              int E) {
  int i = blockIdx.x * blockDim.x + threadIdx.x;
  if (i < E) atomicAdd(&deg[dst[i]], 1.0f);
}

__global__ void rsqrt_inplace(float* __restrict__ deg, int n) {
  int i = blockIdx.x * blockDim.x + threadIdx.x;
  if (i < n) {
    float d = deg[i];
    deg[i] = (d > 0.0f) ? rsqrtf(d) : 0.0f;
  }
}

// out[i][j] = b[j] + h[i][j] * dinv[i]^2   (self-loop term folded in)
__global__ void init_out_selfloop(const float* __restrict__ h,
                                  const float* __restrict__ dinv,
                                  const float* __restrict__ bias,
                                  float* __restrict__ out, int n_elem) {
  int t = blockIdx.x * blockDim.x + threadIdx.x;
  if (t >= n_elem) return;
  int node = t >> 6;
  int j = t & 63;
  float di = dinv[node];
  out[t] = bias[j] + h[t] * di * di;
}

// ---------------------------------------------------------------------------
// Edge scatter: 16 lanes per edge, float4 gather of h[src], 4x atomic f32 adds.
// ---------------------------------------------------------------------------
__global__ void edge_scatter(const float* __restrict__ h,
                             const float* __restrict__ dinv,
                             const int* __restrict__ src,
                             const int* __restrict__ dst,
                             float* __restrict__ out, int E) {
  long long t = (long long)blockIdx.x * blockDim.x + threadIdx.x;
  long long e = t >> 4;
  int q = (int)(t & 15);
  if (e >= E) return;
  int s = src[e];
  int d = dst[e];
  float norm = dinv[s] * dinv[d];
  const float4 hv = *(const float4*)(h + (size_t)s * D + q * 4);  // global_load_b128
  float* o = out + (size_t)d * D + q * 4;
  atomicAdd(o + 0, hv.x * norm);
  atomicAdd(o + 1, hv.y * norm);
  atomicAdd(o + 2, hv.z * norm);
  atomicAdd(o + 3, hv.w * norm);
}

// ---------------------------------------------------------------------------
// Launch
// ---------------------------------------------------------------------------
extern "C" void kernel_launch(void* const* d_in, const int* in_sizes, int n_in,
                              void* d_out, int out_size, void* d_ws, size_t ws_size,
                              hipStream_t stream) {
  const float* x = (const float*)d_in[0];
  const int* ei[3] = {(const int*)d_in[1], (const int*)d_in[2], (const int*)d_in[3]};
  const float* W_ln = (const float*)d_in[4];
  const float* b_ln = (const float*)d_in[5];
  const float* Wk[3] = {(const float*)d_in[6], (const float*)d_in[8],
                        (const float*)d_in[10]};
  const float* bk[3] = {(const float*)d_in[7], (const float*)d_in[9],
                        (const float*)d_in[11]};
  float* out = (float*)d_out;

  const int N = N_NODES;
  const int E = N_EDGES;

  // workspace: h [N*64 f32] | deg/dinv [N f32]
  float* h = (float*)d_ws;
  float* deg = h + (size_t)N * D;

  const int BLK = 256;
  const int gemm_waves = (N / 16) * 4;                 // 25000 tiles
  const int gemm_blocks = (gemm_waves + 7) / 8;        // 8 waves per block
  const int n_blk = (N + BLK - 1) / BLK;
  const int e_blk = (E + BLK - 1) / BLK;
  const int ne_blk = (N * D + BLK - 1) / BLK;
  const long long sc_threads = (long long)E * 16;
  const int sc_blk = (int)((sc_threads + BLK - 1) / BLK);

  // x0 = x @ W_ln + b_ln  -> out[0 .. N*64)
  gemm_wmma_f32<<<gemm_blocks, BLK, 0, stream>>>(x, W_ln, b_ln, out, N);

  for (int k = 0; k < 3; ++k) {
    float* outk = out + (size_t)(k + 1) * N * D;
    const int* srck = ei[k];
    const int* dstk = ei[k] + E;

    // deg = 1 (self loop) + sum over dst
    fill_f32<<<n_blk, BLK, 0, stream>>>(deg, 1.0f, N);
    degree_kernel<<<e_blk, BLK, 0, stream>>>(dstk, deg, E);
    rsqrt_inplace<<<n_blk, BLK, 0, stream>>>(deg, N);   // deg -> dinv in place

    // h = x @ Wk
    gemm_wmma_f32<<<gemm_blocks, BLK, 0, stream>>>(x, Wk[k], nullptr, h, N);

    // out = b + h * dinv^2 (self loop), then scatter real edges
    init_out_selfloop<<<ne_blk, BLK, 0, stream>>>(h, deg, bk[k], outk, N * D);
    edge_scatter<<<sc_blk, BLK, 0, stream>>>(h, deg, srck, dstk, outk, E);
  }
}